// MemristorLinearLayer_71940702208543
// MI455X (gfx1250) — compile-verified
//
#include <hip/hip_runtime.h>

// CDNA5 / gfx1250, wave32.
typedef __attribute__((ext_vector_type(2))) float v2f;
typedef __attribute__((ext_vector_type(8))) float v8f;

#define WAVE_TILE_M 16
#define WAVE_TILE_N 32          // 2 x 16 WMMA tiles per wave -> 512 waves total
#define WAVES_PER_BLOCK 4
#define BLOCK_THREADS (WAVES_PER_BLOCK * 32)
#define KCHUNK 16               // 4 WMMA K-steps per software-pipeline stage

// out[m,n] = sum_k X[m,k] * W[n,k] + bias[n]
// (memristor transform cancels algebraically: out = x @ W^T + bias)
__global__ __launch_bounds__(BLOCK_THREADS)
void memristor_linear_f32_wmma(const float* __restrict__ X,
                               const float* __restrict__ W,
                               const float* __restrict__ bias,
                               float* __restrict__ out,
                               int batch, int size_in, int size_out)
{
    const int lane    = threadIdx.x & 31;
    const int wave    = threadIdx.x >> 5;
    const int waveId  = blockIdx.x * WAVES_PER_BLOCK + wave;

    const int nTilesN = size_out / WAVE_TILE_N;
    const int tileM   = (waveId / nTilesN) * WAVE_TILE_M;
    const int tileN   = (waveId % nTilesN) * WAVE_TILE_N;
    if (tileM >= batch) return;        // wave-uniform: EXEC stays all-ones for WMMA

    const int lane16  = lane & 15;
    const int halfSel = lane >> 4;     // 0: lanes 0-15, 1: lanes 16-31
    const int kShift  = 2 * halfSel;   // A/B fragment K sub-offset per ISA layout

    // Per-lane source rows (contiguous float2 along K each WMMA step).
    const float* arow  = X + (size_t)(tileM + lane16) * size_in + kShift;
    const float* brow0 = W + (size_t)(tileN +  0 + lane16) * size_in + kShift;
    const float* brow1 = W + (size_t)(tileN + 16 + lane16) * size_in + kShift;

    v8f acc0 = {0.f,0.f,0.f,0.f,0.f,0.f,0.f,0.f};
    v8f acc1 = {0.f,0.f,0.f,0.f,0.f,0.f,0.f,0.f};

    // Software pipeline: current chunk in registers, next chunk's 12 loads in
    // flight while the 8 WMMAs of the current chunk execute.
    v2f aC[4], b0C[4], b1C[4];

    #pragma unroll
    for (int s = 0; s < 4; ++s) {
        aC[s]  = *(const v2f*)(arow  + 4 * s);
        b0C[s] = *(const v2f*)(brow0 + 4 * s);
        b1C[s] = *(const v2f*)(brow1 + 4 * s);
    }

    for (int kk = KCHUNK; kk < size_in; kk += KCHUNK) {
        v2f aN[4], b0N[4], b1N[4];
        #pragma unroll
        for (int s = 0; s < 4; ++s) {
            aN[s]  = *(const v2f*)(arow  + kk + 4 * s);
            b0N[s] = *(const v2f*)(brow0 + kk + 4 * s);
            b1N[s] = *(const v2f*)(brow1 + kk + 4 * s);
        }
        #pragma unroll
        for (int s = 0; s < 4; ++s) {
            // 8-arg form: (neg_a, A, neg_b, B, c_mod, C, reuse_a, reuse_b)
            acc0 = __builtin_amdgcn_wmma_f32_16x16x4_f32(false, aC[s], false, b0C[s],
                                                         (short)0, acc0, false, false);
            acc1 = __builtin_amdgcn_wmma_f32_16x16x4_f32(false, aC[s], false, b1C[s],
                                                         (short)0, acc1, false, false);
        }
        #pragma unroll
        for (int s = 0; s < 4; ++s) {
            aC[s] = aN[s]; b0C[s] = b0N[s]; b1C[s] = b1N[s];
        }
    }

    #pragma unroll
    for (int s = 0; s < 4; ++s) {
        acc0 = __builtin_amdgcn_wmma_f32_16x16x4_f32(false, aC[s], false, b0C[s],
                                                     (short)0, acc0, false, false);
        acc1 = __builtin_amdgcn_wmma_f32_16x16x4_f32(false, aC[s], false, b1C[s],
                                                     (short)0, acc1, false, false);
    }

    // D layout: VGPR r -> M = r + 8*halfSel, N = lane16 (per 16x16 tile).
    const int n0  = tileN + lane16;
    const float bn0 = bias[n0];
    const float bn1 = bias[n0 + 16];

    const int rowBase = tileM + 8 * halfSel;
    #pragma unroll
    for (int r = 0; r < 8; ++r) {
        float* orow = out + (size_t)(rowBase + r) * size_out;
        orow[n0]      = acc0[r] + bn0;
        orow[n0 + 16] = acc1[r] + bn1;
    }
}

extern "C" void kernel_launch(void* const* d_in, const int* in_sizes, int n_in,
                              void* d_out, int out_size, void* d_ws, size_t ws_size,
                              hipStream_t stream) {
    const float* X    = (const float*)d_in[0];   // [batch, size_in]
    const float* W    = (const float*)d_in[1];   // [size_out, size_in]
    const float* bias = (const float*)d_in[2];   // [size_out]
    float* out        = (float*)d_out;           // [batch, size_out]

    const int size_out = in_sizes[2];
    const int size_in  = in_sizes[1] / size_out;
    const int batch    = in_sizes[0] / size_in;

    const int wavesTotal = (batch / WAVE_TILE_M) * (size_out / WAVE_TILE_N);
    const int blocks     = wavesTotal / WAVES_PER_BLOCK;

    memristor_linear_f32_wmma<<<blocks, BLOCK_THREADS, 0, stream>>>(
        X, W, bias, out, batch, size_in, size_out);
}